// HungarianLoss_56289841381863
// MI455X (gfx1250) — compile-verified
//
#include <hip/hip_runtime.h>
#include <hip/hip_bf16.h>

// Problem constants (match reference)
#define Bq   8
#define NS   300
#define NT   100
#define CC   512
#define NSQ  300            // square LSA size = max(NS,NT)
#define STILES 19           // ceil(304/16)
#define TTILES 7            // ceil(112/16)
#define LGIOU 10.0f
#define LL1   1.0f

typedef __attribute__((ext_vector_type(2))) float v2f;
typedef __attribute__((ext_vector_type(8))) float v8f;

// ---------------------------------------------------------------------------
// Kernel 1: row-wise log-sum-exp over C=512 logits. One 256-thread block/row.
// ---------------------------------------------------------------------------
__global__ void hl_lse_kernel(const float* __restrict__ logits,
                              float* __restrict__ lse) {
    const int row = blockIdx.x;                 // b*NS + s
    const float* x = logits + (size_t)row * CC;
    const int tid = threadIdx.x;                // 256 threads
    __shared__ float red[256];

    float m = fmaxf(x[tid], x[tid + 256]);
    red[tid] = m; __syncthreads();
    for (int off = 128; off > 0; off >>= 1) {
        if (tid < off) red[tid] = fmaxf(red[tid], red[tid + off]);
        __syncthreads();
    }
    const float mx = red[0];
    __syncthreads();
    float s = expf(x[tid] - mx) + expf(x[tid + 256] - mx);
    red[tid] = s; __syncthreads();
    for (int off = 128; off > 0; off >>= 1) {
        if (tid < off) red[tid] += red[tid + off];
        __syncthreads();
    }
    if (tid == 0) lse[row] = mx + logf(red[0]);
}

// ---------------------------------------------------------------------------
// Kernel 2: pairwise cost via WMMA one-hot GEMM + GIoU/L1 epilogue.
// One wave32 per 16x16 (s,t) output tile.  dot = logits * onehot(tgt)
// using V_WMMA_F32_16X16X4_F32, accumulated over K=512 in steps of 4.
// A 16x4 f32 layout: lanes 0-15 -> M=lane, {v0,v1}={K0,K1};
//                    lanes 16-31 -> M=lane-16, {v0,v1}={K2,K3}.
// B 4x16 f32 layout: v0: lanes0-15 K=0 / lanes16-31 K=2; v1: K=1 / K=3; N=lane&15.
// D 16x16 f32: VGPR i -> M=i (lanes0-15) or M=i+8 (lanes16-31), N=lane&15.
// Exact: every product is 0 or a raw f32 logit; one nonzero per output sum.
// ---------------------------------------------------------------------------
__global__ void hl_cost_kernel(const float* __restrict__ logits,
                               const float* __restrict__ sb,
                               const float* __restrict__ tb,
                               const int*   __restrict__ tgt,
                               const float* __restrict__ lse,
                               float* __restrict__ cost) {
    int tile = blockIdx.x;
    const int tt = tile % TTILES; tile /= TTILES;
    const int st = tile % STILES;
    const int b  = tile / STILES;

    const int lane = threadIdx.x;       // wave32
    const int half = lane >> 4;         // 0: K lo pair, 1: K hi pair
    const int l    = lane & 15;

    // A operand source row (clamped; padded rows discarded at write time)
    const int sA = st * 16 + l;
    const float* arow = logits + ((size_t)b * NS + min(sA, NS - 1)) * CC + half * 2;

    // B operand: one-hot class for my N column (clamped; padded cols discarded)
    const int tcol = tt * 16 + l;
    const int cls  = tgt[b * NT + min(tcol, NT - 1)];

    v8f acc = {};
    for (int k = 0; k < CC; k += 4) {
        v2f a; a.x = arow[k]; a.y = arow[k + 1];
        v2f bb;
        bb.x = (cls == k + half * 2)     ? 1.0f : 0.0f;
        bb.y = (cls == k + half * 2 + 1) ? 1.0f : 0.0f;
        acc = __builtin_amdgcn_wmma_f32_16x16x4_f32(
            /*neg_a=*/false, a, /*neg_b=*/false, bb,
            /*c_mod=*/(short)0, acc, /*reuse_a=*/false, /*reuse_b=*/false);
    }

    // Epilogue: each lane owns 8 (m = i + 8*half, n = l) cells of the tile.
    const int t = tt * 16 + l;
#pragma unroll
    for (int i = 0; i < 8; ++i) {
        const int s = st * 16 + i + 8 * half;
        if (s < NS && t < NT) {
            const float ce = lse[b * NS + s] - acc[i];

            const float* A  = sb + ((size_t)b * NS + s) * 4;
            const float* Bx = tb + ((size_t)b * NT + t) * 4;
            const float ax1 = A[0],  ay1 = A[1],  ax2 = A[2],  ay2 = A[3];
            const float bx1 = Bx[0], by1 = Bx[1], bx2 = Bx[2], by2 = Bx[3];

            const float l1 = 0.25f * (fabsf(ax1 - bx1) + fabsf(ay1 - by1) +
                                      fabsf(ax2 - bx2) + fabsf(ay2 - by2));
            const float ix1 = fmaxf(ax1, bx1), iy1 = fmaxf(ay1, by1);
            const float ix2 = fminf(ax2, bx2), iy2 = fminf(ay2, by2);
            const float inter = fmaxf(ix2 - ix1, 0.0f) * fmaxf(iy2 - iy1, 0.0f);
            const float area_a = (ax2 - ax1) * (ay2 - ay1);
            const float area_b = (bx2 - bx1) * (by2 - by1);
            const float uni = area_a + area_b - inter;
            const float iou = inter / uni;
            const float ex1 = fminf(ax1, bx1), ey1 = fminf(ay1, by1);
            const float ex2 = fmaxf(ax2, bx2), ey2 = fmaxf(ay2, by2);
            const float encl = (ex2 - ex1) * (ey2 - ey1);
            const float giou = iou - (encl - uni) / encl;

            cost[((size_t)b * NS + s) * NT + t] = ce + LGIOU * (1.0f - giou) + LL1 * l1;
        }
    }
}

// ---------------------------------------------------------------------------
// Kernel 3: Jonker-Volgenant LSA, one 512-thread block per batch.
// Cost matrix (300x100 real part) lives in LDS (117 KB; CDNA5 WGP has 320 KB).
// Padded columns j-1 >= NT have implicit cost 0 (matches zero-padded square).
// Dual potentials / minv in f64 to match the reference solver numerics.
// ---------------------------------------------------------------------------
__global__ void hl_jv_kernel(const float* __restrict__ cost,
                             float* __restrict__ out) {
    extern __shared__ unsigned char smem[];
    float*  costl = (float*)smem;                          // NS*NT floats
    double* u     = (double*)(smem + (size_t)NS * NT * 4); // 304 each
    double* v     = u + 304;
    double* minv  = v + 304;
    int*    p     = (int*)(minv + 304);
    int*    way   = p + 304;
    int*    used  = way + 304;

    __shared__ int    sh_j0, sh_j1, sh_i0;
    __shared__ double sh_delta;
    __shared__ double wred[16];
    __shared__ int    wredi[16];

    const int b   = blockIdx.x;
    const int tid = threadIdx.x;
    const int nth = blockDim.x;      // 512 = 16 waves
    const int n   = NSQ;

    const float* cb = cost + (size_t)b * NS * NT;
    for (int i = tid; i < NS * NT; i += nth) costl[i] = cb[i];
    for (int j = tid; j <= n; j += nth) { u[j] = 0.0; v[j] = 0.0; p[j] = 0; way[j] = 0; }
    __syncthreads();

    for (int i = 1; i <= n; ++i) {
        if (tid == 0) { p[0] = i; sh_j0 = 0; }
        for (int j = tid; j <= n; j += nth) { minv[j] = 1e300; used[j] = 0; }
        __syncthreads();

        for (;;) {
            const int j0 = sh_j0;
            if (tid == 0) { used[j0] = 1; sh_i0 = p[j0]; }
            __syncthreads();
            const int i0 = sh_i0;
            const double ui0 = u[i0];

            double bestv = 1e300; int bestj = -1;
            const int j = tid + 1;                  // thread tid owns column j
            if (tid < n && !used[j]) {
                const int c = j - 1;
                const double cur =
                    (double)((c < NT) ? costl[(i0 - 1) * NT + c] : 0.0f) - ui0 - v[j];
                if (cur < minv[j]) { minv[j] = cur; way[j] = j0; }
                bestv = minv[j]; bestj = j;
            }
            // argmin reduce: wave32 shuffle then cross-wave scan (first-index ties)
            for (int off = 16; off > 0; off >>= 1) {
                const double ov = __shfl_down(bestv, off, 32);
                const int    oj = __shfl_down(bestj, off, 32);
                if (ov < bestv) { bestv = ov; bestj = oj; }
            }
            if ((tid & 31) == 0) { wred[tid >> 5] = bestv; wredi[tid >> 5] = bestj; }
            __syncthreads();
            if (tid == 0) {
                double dv = 1e300; int dj = -1;
                for (int w = 0; w < nth / 32; ++w)
                    if (wred[w] < dv) { dv = wred[w]; dj = wredi[w]; }
                sh_delta = dv; sh_j1 = dj;
            }
            __syncthreads();
            const double delta = sh_delta;
            const int    j1    = sh_j1;

            if (tid <= n) {
                if (used[tid]) { u[p[tid]] += delta; v[tid] -= delta; }  // p[] distinct
                else           { minv[tid] -= delta; }
            }
            if (tid == 0) sh_j0 = j1;
            __syncthreads();
            if (p[j1] == 0) break;
        }
        if (tid == 0) {                       // augment along alternating path
            int j0 = sh_j0;
            while (j0) { const int j1 = way[j0]; p[j0] = p[j1]; j0 = j1; }
        }
        __syncthreads();
    }

    // loss = mean over real columns j=1..NT of cost[p[j]-1][j-1]
    double s = 0.0;
    if (tid < NT) s = (double)costl[(p[tid + 1] - 1) * NT + tid];
    for (int off = 16; off > 0; off >>= 1) s += __shfl_down(s, off, 32);
    if ((tid & 31) == 0) wred[tid >> 5] = s;
    __syncthreads();
    if (tid == 0) {
        double tot = 0.0;
        for (int w = 0; w < nth / 32; ++w) tot += wred[w];
        out[b] = (float)(tot / NT);
    }
}

// ---------------------------------------------------------------------------
extern "C" void kernel_launch(void* const* d_in, const int* in_sizes, int n_in,
                              void* d_out, int out_size, void* d_ws, size_t ws_size,
                              hipStream_t stream) {
    const float* logits = (const float*)d_in[0];   // (B, NS, C)
    const float* sb     = (const float*)d_in[1];   // (B, NS, 4)
    const float* tb     = (const float*)d_in[2];   // (B, NT, 4)
    const int*   tgt    = (const int*)d_in[3];     // (B, NT)
    float* out = (float*)d_out;                    // (B,)

    float* lse  = (float*)d_ws;                    // B*NS floats
    float* cost = lse + (size_t)Bq * NS;           // B*NS*NT floats

    hl_lse_kernel<<<Bq * NS, 256, 0, stream>>>(logits, lse);
    hl_cost_kernel<<<Bq * STILES * TTILES, 32, 0, stream>>>(logits, sb, tb, tgt, lse, cost);

    const size_t sh = (size_t)NS * NT * 4 + 3 * 304 * 8 + 3 * 304 * 4; // ~131 KB
    hipFuncSetAttribute((const void*)hl_jv_kernel,
                        hipFuncAttributeMaxDynamicSharedMemorySize, (int)sh);
    hl_jv_kernel<<<Bq, 512, sh, stream>>>(cost, out);
}